// TSApproximation_59373627899945
// MI455X (gfx1250) — compile-verified
//
#include <hip/hip_runtime.h>
#include <hip/hip_bf16.h>

typedef __attribute__((ext_vector_type(2))) float v2f;
typedef __attribute__((ext_vector_type(8))) float v8f;

// out[c, n] = bias[c] + sum_{t<8, d<2} coeff[c,t,d] * pix[n,d]^t
//
// Per wave32, per tile of 16 pixels:
//   D(16x16) = A(16x4) x B(4x16) + C, chained 4x over K (k = 2t + d, k in [0,16)).
//   A rows   = channels (rows 3..15 zero), loop-invariant coefficients.
//   B cols   = 16 pixels; B rows at chain step j are {x^2j, y^2j, x^(2j+1), y^(2j+1)}.
//   C rows   = bias.  Result rows 0..2 (VGPR0..2, lanes 0..15) -> coalesced stores.
__global__ __launch_bounds__(256) void ts_poly_wmma_kernel(
    const float* __restrict__ pix,    // [npix, 2]
    const float* __restrict__ coef,   // [3, 8, 2] flat: c*16 + t*2 + d = c*16 + k
    const float* __restrict__ bias,   // [3]
    float* __restrict__ out,          // [3, npix]
    int npix)
{
    const int  tid   = blockIdx.x * blockDim.x + threadIdx.x;
    const int  lane  = threadIdx.x & 31;
    const int  gwave = tid >> 5;
    const int  nwave = (int)((gridDim.x * blockDim.x) >> 5);
    const bool lower = lane < 16;          // lanes 0-15: K = {4j, 4j+1}; 16-31: K = {4j+2, 4j+3}
    const int  col   = lane & 15;          // A: row M (channel); B/D: column N (pixel in tile)

    // ---- A matrix (coefficients), loaded once per wave ----
    // A 16x4 f32 layout: lane = M; VGPR0 <-> K = 4j + (lower?0:2), VGPR1 <-> K+1.
    v2f a0 = {0.f, 0.f}, a1 = {0.f, 0.f}, a2 = {0.f, 0.f}, a3 = {0.f, 0.f};
    if (col < 3) {
        const float* cr  = coef + col * 16;
        const int    off = lower ? 0 : 2;
        a0[0] = cr[ 0 + off]; a0[1] = cr[ 1 + off];
        a1[0] = cr[ 4 + off]; a1[1] = cr[ 5 + off];
        a2[0] = cr[ 8 + off]; a2[1] = cr[ 9 + off];
        a3[0] = cr[12 + off]; a3[1] = cr[13 + off];
    }
    // C[m,n] = bias[m]: element v of the C/D v8f is row M=v (lower lanes) / M=v+8 (upper).
    const float bias0 = lower ? bias[0] : 0.f;
    const float bias1 = lower ? bias[1] : 0.f;
    const float bias2 = lower ? bias[2] : 0.f;

    const int ntiles = (npix + 15) >> 4;
    const int stride = nwave << 4;
    for (int tile = gwave; tile < ntiles; tile += nwave) {
        const int base = tile << 4;
        const int pidx = base + col;
        const int lidx = pidx < npix ? pidx : (npix - 1);   // clamp tail loads

        v2f p = *(const v2f*)(pix + 2 * lidx);              // coalesced 128B per half-wave

        // Prefetch the next tile this wave will stream (global_prefetch_b8).
        const int nbase = base + stride;                    // wave-uniform condition
        if (nbase < npix) __builtin_prefetch(pix + 2 * (nbase + col), 0, 0);

        const float x  = p[0],  y  = p[1];
        const float x2 = x * x,   y2 = y * y;
        const float x4 = x2 * x2, y4 = y2 * y2;
        const float x6 = x4 * x2, y6 = y4 * y2;
        const float u = lower ? 1.f : x;                    // odd-power factor for upper half
        const float w = lower ? 1.f : y;

        // B 4x16 f32: VGPR0 <-> row K = 4j + (lower?0:2) -> x-power; VGPR1 -> y-power.
        v2f b0, b1, b2, b3;
        b0[0] = u;       b0[1] = w;
        b1[0] = u * x2;  b1[1] = w * y2;
        b2[0] = u * x4;  b2[1] = w * y4;
        b3[0] = u * x6;  b3[1] = w * y6;

        v8f acc = {};
        acc[0] = bias0; acc[1] = bias1; acc[2] = bias2;

        // Chain K = 0..15 in 4 steps of V_WMMA_F32_16X16X4_F32 (EXEC all-1s here).
        acc = __builtin_amdgcn_wmma_f32_16x16x4_f32(false, a0, false, b0, (short)0, acc, false, false);
        acc = __builtin_amdgcn_wmma_f32_16x16x4_f32(false, a1, false, b1, (short)0, acc, false, false);
        acc = __builtin_amdgcn_wmma_f32_16x16x4_f32(false, a2, false, b2, (short)0, acc, false, false);
        acc = __builtin_amdgcn_wmma_f32_16x16x4_f32(false, a3, false, b3, (short)0, acc, false, false);

        // D row c, columns 0..15 live in acc[c] of lanes 0..15 -> 64B coalesced stores.
        if (lower && pidx < npix) {
            out[pidx]            = acc[0];
            out[npix + pidx]     = acc[1];
            out[2 * npix + pidx] = acc[2];
        }
    }
}

extern "C" void kernel_launch(void* const* d_in, const int* in_sizes, int n_in,
                              void* d_out, int out_size, void* d_ws, size_t ws_size,
                              hipStream_t stream) {
    const float* pix  = (const float*)d_in[0];   // [N, 2] f32
    const float* coef = (const float*)d_in[1];   // [3, 8, 2] f32
    const float* bias = (const float*)d_in[2];   // [3] f32
    float* out = (float*)d_out;                  // [3, N] f32

    const int npix  = in_sizes[0] / 2;
    const int block = 256;                       // 8 wave32s per block
    const int ntiles = (npix + 15) >> 4;
    int grid = 4096;                             // 32K waves -> ~8 tiles/wave, HBM-saturating
    const int maxGrid = (ntiles + (block / 32) - 1) / (block / 32);
    if (grid > maxGrid) grid = maxGrid;
    if (grid < 1) grid = 1;

    ts_poly_wmma_kernel<<<grid, block, 0, stream>>>(pix, coef, bias, out, npix);
}